// GNN_Node_42657615183923
// MI455X (gfx1250) — compile-verified
//
#include <hip/hip_runtime.h>
#include <hip/hip_bf16.h>

typedef __attribute__((ext_vector_type(2))) float v2f;
typedef __attribute__((ext_vector_type(8))) float v8f;

#define BN_EPS 1e-5f

// ---------------------------------------------------------------- utilities
__global__ void gnn_zero_f32(float* __restrict__ p, int n) {
    int i = blockIdx.x * blockDim.x + threadIdx.x;
    if (i < n) p[i] = 0.0f;
}

// one thread per edge: deg[dst] += 1
__global__ void gnn_deg_count(const int* __restrict__ dst, float* __restrict__ deg, int E) {
    int e = blockIdx.x * blockDim.x + threadIdx.x;
    if (e < E) atomicAdd(&deg[dst[e]], 1.0f);
}

__global__ void gnn_inv_deg(const float* __restrict__ deg, float* __restrict__ invd, int N) {
    int i = blockIdx.x * blockDim.x + threadIdx.x;
    if (i < N) invd[i] = 1.0f / fmaxf(deg[i], 1.0f);
}

// AtomEncoder: h[n,d] = sum_f emb[f, x[n,f], d]; also initialize out = h
__global__ void gnn_embed(const int* __restrict__ x, const float* __restrict__ emb,
                          float* __restrict__ h, float* __restrict__ out, int N, int V) {
    int i = blockIdx.x * blockDim.x + threadIdx.x;
    if (i >= N * 64) return;
    int n = i >> 6, d = i & 63;
    const int* xr = x + n * 9;
    float s = 0.0f;
#pragma unroll
    for (int f = 0; f < 9; ++f) {
        int idx = xr[f];
        s += emb[((size_t)(f * V + idx)) * 64 + d];
    }
    h[i] = s;
    out[i] = s;
}

// neighbor scatter-add: agg[dst,:] += h[src,:]; one thread per (edge, 4 channels)
__global__ void gnn_scatter(const int* __restrict__ ei, const float* __restrict__ h,
                            float* __restrict__ agg, int E) {
    int j = blockIdx.x * blockDim.x + threadIdx.x;
    if (j >= E * 16) return;
    int e = j >> 4;
    int q = (j & 15) << 2;
    int s = ei[e];
    int d = ei[E + e];
    const float4 v = *(const float4*)(h + (size_t)s * 64 + q);
    float* a = agg + (size_t)d * 64 + q;
    atomicAdd(a + 0, v.x);
    atomicAdd(a + 1, v.y);
    atomicAdd(a + 2, v.z);
    atomicAdd(a + 3, v.w);
}

// -------------------------------------------------- SAGE GEMM + BN statistics
// hn = (agg * inv_deg) @ Wl + bl + h @ Wr, plus per-channel sum / sumsq.
// 128 threads = 4 waves; each wave owns a 16-row tile (block covers 64 rows).
// fp32 WMMA: 16 k-steps (K=4) x 4 column tiles x 2 weight matrices.
__global__ __launch_bounds__(128) void gnn_sage_gemm_stats(
    const float* __restrict__ agg, const float* __restrict__ invd,
    const float* __restrict__ h, const float* __restrict__ Wl,
    const float* __restrict__ Wr, const float* __restrict__ bl,
    float* __restrict__ hn, float* __restrict__ gsum, float* __restrict__ gsq, int N) {
    __shared__ float sWl[64 * 64];
    __shared__ float sWr[64 * 64];
    __shared__ float ssum[64];
    __shared__ float ssq[64];

    const int tid = threadIdx.x;
    for (int i = tid; i < 1024; i += 128) {
        ((float4*)sWl)[i] = ((const float4*)Wl)[i];
        ((float4*)sWr)[i] = ((const float4*)Wr)[i];
    }
    if (tid < 64) { ssum[tid] = 0.0f; ssq[tid] = 0.0f; }
    __syncthreads();

    const int lane  = tid & 31;
    const int wave  = tid >> 5;
    const int r16   = lane & 15;   // A row / B,C column within tile
    const int khalf = lane >> 4;   // selects K pair {0,1} vs {2,3}
    const int rowBase = blockIdx.x * 64 + wave * 16;

    int row = rowBase + r16;
    if (row > N - 1) row = N - 1;  // clamp loads; stores/stats are masked below
    const float sc = invd[row];
    const float* aggp = agg + (size_t)row * 64;
    const float* hp   = h + (size_t)row * 64;

    v8f acc[4] = {v8f{}, v8f{}, v8f{}, v8f{}};

#pragma unroll
    for (int k = 0; k < 64; k += 4) {
        const int kk = k + khalf * 2;
        // A fragments (16x4): lane r16 holds row, components = K kk, kk+1
        float2 al = *(const float2*)(aggp + kk);
        float2 ar = *(const float2*)(hp + kk);
        v2f aL; aL.x = al.x * sc; aL.y = al.y * sc;
        v2f aR; aR.x = ar.x;      aR.y = ar.y;

        const float* wl0 = sWl + kk * 64 + r16;
        const float* wl1 = sWl + (kk + 1) * 64 + r16;
        const float* wr0 = sWr + kk * 64 + r16;
        const float* wr1 = sWr + (kk + 1) * 64 + r16;

#pragma unroll
        for (int t = 0; t < 4; ++t) {
            // B fragments (4x16): lane r16 holds column t*16+r16, comps = K kk, kk+1
            v2f bL; bL.x = wl0[t * 16]; bL.y = wl1[t * 16];
            v2f bR; bR.x = wr0[t * 16]; bR.y = wr1[t * 16];
            acc[t] = __builtin_amdgcn_wmma_f32_16x16x4_f32(
                false, aL, false, bL, (short)0, acc[t], false, false);
            acc[t] = __builtin_amdgcn_wmma_f32_16x16x4_f32(
                false, aR, false, bR, (short)0, acc[t], false, false);
        }
    }

    // bias + store + per-channel partial stats (C layout: VGPR r -> row r/khalf*8+r)
#pragma unroll
    for (int t = 0; t < 4; ++t) {
        const int col = t * 16 + r16;
        const float bias = bl[col];
        float ps = 0.0f, pq = 0.0f;
#pragma unroll
        for (int r = 0; r < 8; ++r) {
            const int orow = rowBase + khalf * 8 + r;
            const float v = acc[t][r] + bias;
            if (orow < N) {
                hn[(size_t)orow * 64 + col] = v;
                ps += v;
                pq += v * v;
            }
        }
        atomicAdd(&ssum[col], ps);  // LDS float atomics (ds_add_f32)
        atomicAdd(&ssq[col], pq);
    }
    __syncthreads();
    if (tid < 64) {
        atomicAdd(&gsum[tid], ssum[tid]);
        atomicAdd(&gsq[tid], ssq[tid]);
    }
}

// fold BN into per-channel scale/shift
__global__ void gnn_bn_prep(const float* __restrict__ gsum, const float* __restrict__ gsq,
                            const float* __restrict__ gamma, const float* __restrict__ beta,
                            float* __restrict__ scale, float* __restrict__ shift, float invN) {
    int c = threadIdx.x;  // 64 threads
    float mu  = gsum[c] * invN;
    float var = gsq[c] * invN - mu * mu;
    float rstd = rsqrtf(var + BN_EPS);
    float s = gamma[c] * rstd;
    scale[c] = s;
    shift[c] = beta[c] - mu * s;
}

// hn = BN(hn) [+ReLU]; out += hn; h = hn
__global__ void gnn_bn_apply(const float* __restrict__ hn, const float* __restrict__ scale,
                             const float* __restrict__ shift, float* __restrict__ h,
                             float* __restrict__ out, int N, int relu) {
    int i = blockIdx.x * blockDim.x + threadIdx.x;
    if (i >= N * 64) return;
    int c = i & 63;
    float v = hn[i] * scale[c] + shift[c];
    if (relu) v = fmaxf(v, 0.0f);
    out[i] += v;
    h[i] = v;
}

// ---------------------------------------------------------------- launcher
extern "C" void kernel_launch(void* const* d_in, const int* in_sizes, int n_in,
                              void* d_out, int out_size, void* d_ws, size_t ws_size,
                              hipStream_t stream) {
    const int* x      = (const int*)d_in[0];
    const int* ei     = (const int*)d_in[1];
    /* d_in[2] batch: unused */
    const float* emb  = (const float*)d_in[3];
    const float* Wl   = (const float*)d_in[4];
    const float* bl   = (const float*)d_in[5];
    const float* Wr   = (const float*)d_in[6];
    const float* gamma= (const float*)d_in[7];
    const float* beta = (const float*)d_in[8];
    float* out = (float*)d_out;

    const int NF = 9, D = 64, L = 5;
    const int N = in_sizes[0] / NF;
    const int E = in_sizes[1] / 2;
    const int V = in_sizes[3] / (NF * D);
    const int ND = N * D;

    float* ws    = (float*)d_ws;
    float* h     = ws; ws += (size_t)ND;
    float* hn    = ws; ws += (size_t)ND;
    float* agg   = ws; ws += (size_t)ND;
    float* deg   = ws; ws += (size_t)N;
    float* invd  = ws; ws += (size_t)N;
    float* gsum  = ws; ws += 64;   // gsum & gsq contiguous: zeroed together
    float* gsq   = ws; ws += 64;
    float* bscal = ws; ws += 64;
    float* bshft = ws; ws += 64;

    const int T = 256;
    gnn_zero_f32<<<(N + T - 1) / T, T, 0, stream>>>(deg, N);
    gnn_deg_count<<<(E + T - 1) / T, T, 0, stream>>>(ei + E, deg, E);
    gnn_inv_deg<<<(N + T - 1) / T, T, 0, stream>>>(deg, invd, N);
    gnn_embed<<<(ND + T - 1) / T, T, 0, stream>>>(x, emb, h, out, N, V);

    for (int l = 0; l < L; ++l) {
        gnn_zero_f32<<<(ND + T - 1) / T, T, 0, stream>>>(agg, ND);
        gnn_zero_f32<<<1, 128, 0, stream>>>(gsum, 128);
        gnn_scatter<<<(E * 16 + T - 1) / T, T, 0, stream>>>(ei, h, agg, E);
        gnn_sage_gemm_stats<<<(N + 63) / 64, 128, 0, stream>>>(
            agg, invd, h, Wl + (size_t)l * D * D, Wr + (size_t)l * D * D,
            bl + (size_t)l * D, hn, gsum, gsq, N);
        gnn_bn_prep<<<1, 64, 0, stream>>>(gsum, gsq, gamma + (size_t)l * D,
                                          beta + (size_t)l * D, bscal, bshft,
                                          1.0f / (float)N);
        gnn_bn_apply<<<(ND + T - 1) / T, T, 0, stream>>>(hn, bscal, bshft, h, out, N,
                                                         (l < L - 1) ? 1 : 0);
    }
}